// SoftAssignPointLoss_9887014716139
// MI455X (gfx1250) — compile-verified
//
#include <hip/hip_runtime.h>
#include <math.h>

#define SIGMA 2.0f
#define EPSV  1e-8f
#define Nb 4
#define Pp 64
#define Zd 32
#define Yd 128
#define Xd 128

typedef float v2f __attribute__((ext_vector_type(2)));
typedef float v8f __attribute__((ext_vector_type(8)));

// Workspace layout (floats):
//   Kz: [N][P][Z]   Ky: [N][P][Y]   Kx: [N][P][X]   m: [N][P]
#define KZ_OFF 0
#define KY_OFF (Nb*Pp*Zd)
#define KX_OFF (KY_OFF + Nb*Pp*Yd)
#define M_OFF  (KX_OFF + Nb*Pp*Xd)

// Skewed LDS index: element (y,x) lives at y*Xd + ((x+y) & (Xd-1)).
// Row stride 128 floats => bank(addr) = (x+y) mod 64 for both the pass-A
// row-writes (x varies) and the pass-B column-reads (y varies): conflict-free
// on the 64-bank LDS without exceeding 64KB static allocation.
#define WIDX(y, x) ((y) * Xd + (((x) + (y)) & (Xd - 1)))

// ---------------------------------------------------------------------------
// Kernel 1: per-(n,p) separable Gaussian tables, and zero the m accumulator.
// grid = N*P blocks, 128 threads.
// ---------------------------------------------------------------------------
__global__ void sapl_precompute(const float* __restrict__ pts, float* __restrict__ ws) {
    const int np  = blockIdx.x;          // n*P + p
    const int tid = threadIdx.x;         // 0..127
    const float pz = pts[np * 3 + 0];
    const float py = pts[np * 3 + 1];
    const float px = pts[np * 3 + 2];
    const float c = -1.0f / (2.0f * SIGMA * SIGMA);

    if (tid < Zd) { float d = (float)tid - pz; ws[KZ_OFF + np * Zd + tid] = __expf(d * d * c); }
    { float d = (float)tid - py; ws[KY_OFF + np * Yd + tid] = __expf(d * d * c); }
    { float d = (float)tid - px; ws[KX_OFF + np * Xd + tid] = __expf(d * d * c); }
    if (tid == 0) ws[M_OFF + np] = 0.0f;
}

// ---------------------------------------------------------------------------
// Kernel 2: one block per (n,z) slice. 256 threads = 8 wave32.
// Pass A: denom = [Kz*Ky](YxP) @ Kx(PxX) via v_wmma_f32_16x16x4_f32,
//         then w = sigmoid(logit)/max(denom,eps) into skewed LDS.
// Pass B: T(PxY) = Kx(PxX) @ w^T(XxY) via WMMA, scale by Kz*Ky, reduce into m.
// ---------------------------------------------------------------------------
__global__ void __launch_bounds__(256) sapl_slice(const float* __restrict__ logits,
                                                  float* __restrict__ ws) {
    __shared__ float w_s[Yd * Xd];  // 64 KB, skewed layout (see WIDX)

    const int nz   = blockIdx.x;    // n*Z + z
    const int n    = nz / Zd;
    const int z    = nz % Zd;
    const int lane = threadIdx.x & 31;
    const int wave = threadIdx.x >> 5;
    const int half = lane >> 4;     // 0 for lanes 0-15, 1 for lanes 16-31
    const int l15  = lane & 15;

    const float* __restrict__ Kz = ws + KZ_OFF + n * Pp * Zd;
    const float* __restrict__ Ky = ws + KY_OFF + n * Pp * Yd;
    const float* __restrict__ Kx = ws + KX_OFF + n * Pp * Xd;
    const float* __restrict__ lg = logits + ((size_t)n * Zd + z) * (size_t)(Yd * Xd);
    float* __restrict__ m = ws + M_OFF + n * Pp;

    // ---------------- Pass A: denom tiles + w ----------------
    // 8x8 = 64 output tiles of 16x16; each of the 8 waves owns 8 tiles.
    for (int t = wave; t < (Yd / 16) * (Xd / 16); t += 8) {
        const int y0 = (t >> 3) * 16;
        const int x0 = (t & 7) * 16;
        const int yA = y0 + l15;     // A-fragment row (M)
        const int xB = x0 + l15;     // B-fragment col (N)

        v8f acc = {};
#pragma unroll
        for (int kb = 0; kb < Pp; kb += 4) {       // K = point index p
            const int p0 = kb + half * 2;
            v2f a, b;
            a.x = Kz[p0 * Zd + z]       * Ky[p0 * Yd + yA];
            a.y = Kz[(p0 + 1) * Zd + z] * Ky[(p0 + 1) * Yd + yA];
            b.x = Kx[p0 * Xd + xB];
            b.y = Kx[(p0 + 1) * Xd + xB];
            acc = __builtin_amdgcn_wmma_f32_16x16x4_f32(
                false, a, false, b, (short)0, acc, false, false);
        }

        // w = sigmoid(logit) / max(denom, eps)
#pragma unroll
        for (int i = 0; i < 8; ++i) {
            const int y = y0 + i + half * 8;
            const int x = x0 + l15;
            const float lv = lg[y * Xd + x];
            const float s  = 1.0f / (1.0f + __expf(-lv));
            w_s[WIDX(y, x)] = s / fmaxf(acc[i], EPSV);
        }
    }

    __syncthreads();

    // ---------------- Pass B: T = Kx @ w^T, fold into m ----------------
    // 4x8 = 32 output tiles (P x Y); each wave owns 4 tiles.
    for (int t = wave; t < (Pp / 16) * (Yd / 16); t += 8) {
        const int p0 = (t >> 3) * 16;
        const int y0 = (t & 7) * 16;
        const int pA = p0 + l15;     // A-fragment row (M = p)
        const int yB = y0 + l15;     // B-fragment col (N = y)

        v8f acc = {};
#pragma unroll
        for (int kb = 0; kb < Xd; kb += 4) {       // K = x coordinate
            const int xk = kb + half * 2;
            v2f a, b;
            a.x = Kx[pA * Xd + xk];
            a.y = Kx[pA * Xd + xk + 1];
            b.x = w_s[WIDX(yB, xk)];                // B[K=x][N=y] = w[y][x]
            b.y = w_s[WIDX(yB, xk + 1)];
            acc = __builtin_amdgcn_wmma_f32_16x16x4_f32(
                false, a, false, b, (short)0, acc, false, false);
        }

        // m[p] += Kz[p,z] * sum_y Ky[p,y] * T[p,y]
#pragma unroll
        for (int i = 0; i < 8; ++i) {
            const int p = p0 + i + half * 8;
            const int y = y0 + l15;
            float v = acc[i] * Kz[p * Zd + z] * Ky[p * Yd + y];
            // reduce across the 16 lanes of this half (same p, different y)
            v += __shfl_xor(v, 1, 32);
            v += __shfl_xor(v, 2, 32);
            v += __shfl_xor(v, 4, 32);
            v += __shfl_xor(v, 8, 32);
            if (l15 == 0) atomicAdd(&m[p], v);
        }
    }
}

// ---------------------------------------------------------------------------
// Kernel 3: loss = mean_n mean_p ( -log(max(m, eps)) )  ->  scalar
// ---------------------------------------------------------------------------
__global__ void sapl_finalize(const float* __restrict__ ws, float* __restrict__ out) {
    __shared__ float red[Nb * Pp];
    const int tid = threadIdx.x;                 // 0..255 == one (n,p) each
    red[tid] = -__logf(fmaxf(ws[M_OFF + tid], EPSV));
    __syncthreads();
    for (int s = (Nb * Pp) / 2; s > 0; s >>= 1) {
        if (tid < s) red[tid] += red[tid + s];
        __syncthreads();
    }
    if (tid == 0) out[0] = red[0] / (float)(Nb * Pp);
}

extern "C" void kernel_launch(void* const* d_in, const int* in_sizes, int n_in,
                              void* d_out, int out_size, void* d_ws, size_t ws_size,
                              hipStream_t stream) {
    const float* logits = (const float*)d_in[0];   // [N,1,Z,Y,X] f32
    const float* pts    = (const float*)d_in[1];   // [N,P,3]     f32
    float* ws  = (float*)d_ws;
    float* out = (float*)d_out;

    sapl_precompute<<<Nb * Pp, 128, 0, stream>>>(pts, ws);
    sapl_slice<<<Nb * Zd, 256, 0, stream>>>(logits, ws);
    sapl_finalize<<<1, Nb * Pp, 0, stream>>>(ws, out);
}